// CostModule_18906446037686
// MI455X (gfx1250) — compile-verified
//
#include <hip/hip_runtime.h>
#include <stdint.h>

// Problem constants (fixed by the reference)
#define B_    64
#define N_    600
#define R_    40
#define L_    50
#define NN    (N_ * N_)        // 360000 elements per batch image
#define GB    88               // stage-1 blocks per batch
#define CHUNK 4096             // elements per stage-1 block (88*4096 >= 360000)

// Output tuple offsets (flat, reference return order)
#define OFF_TDT   0                       // total_demand_time   [B]
#define OFF_TRT   (OFF_TDT + B_)          // total_route_time    [B]
#define OFF_TAT   (OFF_TRT + B_)          // trips_at_transfers  [B,4]
#define OFF_TD    (OFF_TAT + B_ * 4)      // total_demand        [B]
#define OFF_UD    (OFF_TD + B_)           // unserved_demand     [B]
#define OFF_TTR   (OFF_UD + B_)           // total_transfers     [B]
#define OFF_TRIP  (OFF_TTR + B_)          // trip_times          [B,N,N]
#define OFF_OOB   (OFF_TRIP + B_ * NN)    // n_stops_oob         [B]
#define OFF_NSV   (OFF_OOB + B_)          // n_stops_visited     [B,R]

typedef float v2f __attribute__((ext_vector_type(2)));
typedef float v8f __attribute__((ext_vector_type(8)));

// ---------------------------------------------------------------------------
// Stage 1: bandwidth-bound streaming pass. Each block handles one contiguous
// 4096-element chunk of one batch image; 128-bit loads/stores per lane.
// Produces trip_times and a 16-float partial record (8 scalars + 8 zeros).
// ---------------------------------------------------------------------------
__global__ __launch_bounds__(256) void stage1_kernel(
    const float* __restrict__ demand, const float* __restrict__ transit,
    const int* __restrict__ ntrans, const unsigned char* __restrict__ haspath,
    float* __restrict__ out, float* __restrict__ ws)
{
    const int blk = blockIdx.x;
    const int b   = blk / GB;
    const int gb  = blk - b * GB;
    const int tid = threadIdx.x;
    const int base = b * NN;

    float a_tdt = 0.f, a_ttr = 0.f, a_ud = 0.f, a_td = 0.f;
    float a_t0 = 0.f, a_t1 = 0.f, a_t2 = 0.f, a_t3 = 0.f;

    auto elem = [&](float d, float t, int n, unsigned char h, float& tt) {
        const bool hp = (h != 0);
        tt = hp ? t : 0.f;
        a_tdt = fmaf(d, tt, a_tdt);
        a_ttr = fmaf(d, (float)n, a_ttr);
        a_td += d;
        a_ud += hp ? 0.f : d;
        const int bin = hp ? (n > 2 ? 3 : n) : 3;   // no-path counted as 3 transfers
        a_t0 += (bin == 0) ? d : 0.f;
        a_t1 += (bin == 1) ? d : 0.f;
        a_t2 += (bin == 2) ? d : 0.f;
        a_t3 += (bin == 3) ? d : 0.f;
    };

#pragma unroll
    for (int it = 0; it < 4; ++it) {
        const int local = gb * CHUNK + (it * 256 + tid) * 4;
        if (local < NN) {                      // NN % 4 == 0: groups fully in/out
            const int idx = base + local;
            const float4 dv = *reinterpret_cast<const float4*>(demand + idx);
            const float4 tv = *reinterpret_cast<const float4*>(transit + idx);
            const int4   nv = *reinterpret_cast<const int4*>(ntrans + idx);
            const uchar4 hv = *reinterpret_cast<const uchar4*>(haspath + idx);
            float4 ttv;
            elem(dv.x, tv.x, nv.x, hv.x, ttv.x);
            elem(dv.y, tv.y, nv.y, hv.y, ttv.y);
            elem(dv.z, tv.z, nv.z, hv.z, ttv.z);
            elem(dv.w, tv.w, nv.w, hv.w, ttv.w);
            *reinterpret_cast<float4*>(out + OFF_TRIP + idx) = ttv;
        }
    }

    // Deterministic wave32 reduction, then fixed-order cross-wave LDS sum.
    float acc[8] = {a_tdt, a_ttr, a_ud, a_td, a_t0, a_t1, a_t2, a_t3};
#pragma unroll
    for (int s = 0; s < 8; ++s) {
#pragma unroll
        for (int off = 16; off >= 1; off >>= 1)
            acc[s] += __shfl_xor(acc[s], off, 32);
    }

    __shared__ float sred[8][8];               // [wave][scalar]
    const int lane = tid & 31, wv = tid >> 5;
    if (lane == 0) {
#pragma unroll
        for (int s = 0; s < 8; ++s) sred[wv][s] = acc[s];
    }
    __syncthreads();

    float* wp = ws + (size_t)(b * GB + gb) * 16;
    if (tid < 8) {
        float tot = 0.f;
#pragma unroll
        for (int w = 0; w < 8; ++w) tot += sred[w][tid];
        wp[tid] = tot;
    } else if (tid < 16) {
        wp[tid] = 0.f;                         // keep WMMA columns 8..15 zero
    }
}

// ---------------------------------------------------------------------------
// Stage 2: per-batch reduction of GB partial records using
// V_WMMA_F32_16X16X4_F32 with an all-ones A matrix:
//   D[m][n] = sum_k B[k][n] + C[m][n]  -> each WMMA folds 4 partial vectors.
// Column n = lane%16 carries scalar n; the sum over the 4 K-rows is invariant
// to the K<->VGPR/lane-half mapping, so the result is layout-robust and exact.
// One fully-active wave32 per block (EXEC all-ones as WMMA requires).
// ---------------------------------------------------------------------------
__global__ __launch_bounds__(32) void stage2_kernel(
    const float* __restrict__ ws, float* __restrict__ out)
{
    const int b    = blockIdx.x;
    const int lane = threadIdx.x;
    const int col  = lane & 15;
    const int hi   = lane >> 4;                // 0: K rows {0,1}; 1: K rows {2,3}
    const float* P = ws + (size_t)b * GB * 16;

    v8f acc = {};
    v2f av;  av.x = 1.0f;  av.y = 1.0f;        // A = ones(16x4)

    for (int g = 0; g < GB; g += 4) {
        v2f bv;
        bv.x = P[(g + (hi ? 2 : 0)) * 16 + col];
        bv.y = P[(g + (hi ? 3 : 1)) * 16 + col];
        acc = __builtin_amdgcn_wmma_f32_16x16x4_f32(
            /*neg_a=*/false, av, /*neg_b=*/false, bv,
            /*c_mod=*/(short)0, acc, /*reuse_a=*/false, /*reuse_b=*/false);
    }

    const float r = acc[0];                    // D[0][col] (same for all rows)
    if (lane == 0)      out[OFF_TDT + b] = r;                 // total_demand_time
    else if (lane == 1) out[OFF_TTR + b] = r;                 // total_transfers
    else if (lane == 2) out[OFF_UD  + b] = r;                 // unserved_demand
    else if (lane == 3) out[OFF_TD  + b] = r;                 // total_demand
    else if (lane < 8)  out[OFF_TAT + b * 4 + (lane - 4)] = r; // histogram bins
}

// ---------------------------------------------------------------------------
// Routes: per-route length / distinct-stop count (600-bit bitmask), length
// penalties reduced in fixed order, plus total_route_time passthrough.
// ---------------------------------------------------------------------------
__global__ __launch_bounds__(64) void routes_kernel(
    const int* __restrict__ routes, const float* __restrict__ trt,
    const int* __restrict__ nleft, const unsigned char* __restrict__ hascur,
    float* __restrict__ out)
{
    __shared__ float sdelta[64];
    const int b = blockIdx.x, t = threadIdx.x;
    float delta = 0.f;

    if (t < R_) {
        const int* rt = routes + (b * R_ + t) * L_;
        unsigned mask[(N_ + 31) / 32];
#pragma unroll
        for (int w = 0; w < (N_ + 31) / 32; ++w) mask[w] = 0u;
        int len = 0, distinct = 0;
        for (int i = 0; i < L_; ++i) {
            const int si = rt[i];
            if (si > -1) {
                ++len;
                const unsigned w = (unsigned)si >> 5, bit = 1u << (si & 31);
                if (!(mask[w] & bit)) { ++distinct; mask[w] |= bit; }
            }
        }
        int d1 = 2 - len;  if (d1 < 0) d1 = 0;  if (len == 0) d1 = 0;
        int d2 = len - 48; if (d2 < 0) d2 = 0;
        delta = (float)(d1 + d2);
        out[OFF_NSV + b * R_ + t] = (float)distinct;
    }
    sdelta[t] = delta;
    __syncthreads();

    if (t == 0) {
        float sum = 0.f;
        for (int i = 0; i < 64; ++i) sum += sdelta[i];
        const float nun = (float)nleft[b] - (hascur[b] ? 1.f : 0.f);
        out[OFF_OOB + b] = sum + nun * 2.0f;   // MIN_ROUTE_LEN = 2
        out[OFF_TRT + b] = trt[b];             // passthrough output
    }
}

extern "C" void kernel_launch(void* const* d_in, const int* in_sizes, int n_in,
                              void* d_out, int out_size, void* d_ws, size_t ws_size,
                              hipStream_t stream)
{
    const float*         demand  = (const float*)d_in[0];
    const float*         transit = (const float*)d_in[1];
    const float*         trt     = (const float*)d_in[2];
    const int*           routes  = (const int*)d_in[3];
    const int*           ntrans  = (const int*)d_in[4];
    const unsigned char* haspath = (const unsigned char*)d_in[5];
    const int*           nleft   = (const int*)d_in[6];
    const unsigned char* hascur  = (const unsigned char*)d_in[7];
    // d_in[8] (n_disconnected_demand_edges) is unused by the reference outputs.

    float* out = (float*)d_out;
    float* ws  = (float*)d_ws;   // needs B*GB*16 floats = 360,448 bytes

    stage1_kernel<<<dim3(B_ * GB), dim3(256), 0, stream>>>(
        demand, transit, ntrans, haspath, out, ws);
    stage2_kernel<<<dim3(B_), dim3(32), 0, stream>>>(ws, out);
    routes_kernel<<<dim3(B_), dim3(64), 0, stream>>>(routes, trt, nleft, hascur, out);
}